// GraphAttentionLayer_47236050321750
// MI455X (gfx1250) — compile-verified
//
#include <hip/hip_runtime.h>
#include <hip/hip_bf16.h>

#define IN_F 128
#define HID 128
#define NCTX 32
#define ALPHA 0.2f
#define NEG_INF -9000000000000000.0f

typedef __attribute__((ext_vector_type(2))) float v2f;
typedef __attribute__((ext_vector_type(8))) float v8f;

// ---------------------------------------------------------------------------
// Kernel 1: u = W_i @ a_i, v = W_j @ a_j   (128x128 each, trivial)
// ---------------------------------------------------------------------------
__global__ __launch_bounds__(128) void gat_uv(const float* __restrict__ Wi,
                                              const float* __restrict__ Wj,
                                              const float* __restrict__ aw,
                                              float* __restrict__ u,
                                              float* __restrict__ v) {
    int f = threadIdx.x;  // 0..127
    float su = 0.f, sv = 0.f;
#pragma unroll 8
    for (int k = 0; k < HID; ++k) {
        su = fmaf(Wi[f * HID + k], aw[k], su);
        sv = fmaf(Wj[f * HID + k], aw[HID + k], sv);
    }
    u[f] = su;
    v[f] = sv;
}

// ---------------------------------------------------------------------------
// Kernel 2: per-node scores. One wave32 per node.
//   s_i[n] = h[n].u + a_b    (self term, bias folded in)
//   s_all[n] = h[n].v        (neighbor term, gathered later)
// ---------------------------------------------------------------------------
__global__ __launch_bounds__(256) void gat_scores(const float* __restrict__ h,
                                                  const float* __restrict__ u,
                                                  const float* __restrict__ v,
                                                  const float* __restrict__ ab,
                                                  float* __restrict__ s_i,
                                                  float* __restrict__ s_all,
                                                  int N) {
    int node = (int)((blockIdx.x * (unsigned)blockDim.x + threadIdx.x) >> 5);
    int lane = threadIdx.x & 31;
    if (node >= N) return;

    const float* hr = h + (size_t)node * IN_F + lane * 4;
    float h0 = hr[0], h1 = hr[1], h2 = hr[2], h3 = hr[3];
    const float* up = u + lane * 4;
    const float* vp = v + lane * 4;
    float di = h0 * up[0] + h1 * up[1] + h2 * up[2] + h3 * up[3];
    float dj = h0 * vp[0] + h1 * vp[1] + h2 * vp[2] + h3 * vp[3];
#pragma unroll
    for (int off = 16; off > 0; off >>= 1) {
        di += __shfl_xor(di, off, 32);
        dj += __shfl_xor(dj, off, 32);
    }
    if (lane == 0) {
        s_i[node] = di + ab[0];
        s_all[node] = dj;
    }
}

// ---------------------------------------------------------------------------
// Kernel 3: main. 16 nodes per block of 256 threads (8 wave32).
//  Phase 1: softmax over 32 context scores per node (one wave handles 2
//           nodes), attn-weighted gather-accumulate of raw h rows -> LDS.
//  Phase 2: out[16 x 128] = mixed[16 x 128] @ W_j[128 x 128] using
//           V_WMMA_F32_16X16X4_F32; wave w owns output columns [16w,16w+16).
// ---------------------------------------------------------------------------
__global__ __launch_bounds__(256) void gat_main(const float* __restrict__ h,
                                                const long long* __restrict__ idx,
                                                const float* __restrict__ Wj,
                                                const float* __restrict__ s_i,
                                                const float* __restrict__ s_all,
                                                float* __restrict__ out,
                                                int N) {
    __shared__ float mixed[16][132];   // stride 132 -> conflict-free column reads
    __shared__ int   nbr[16][NCTX];

    const int tile = blockIdx.x;
    const int lane = threadIdx.x & 31;
    const int wave = threadIdx.x >> 5;

    // ---- Phase 1: softmax + mix raw h rows -------------------------------
#pragma unroll
    for (int r = 0; r < 2; ++r) {
        const int row = wave * 2 + r;           // 0..15
        const int n = tile * 16 + row;
        const bool live = (n < N);

        long long id = -1;
        if (live) id = idx[(size_t)n * NCTX + lane];
        const int jd = (id >= 0) ? (int)id : -1;
        nbr[row][lane] = jd;

        float e;
        if (jd >= 0) {
            float t = s_i[n] + s_all[jd];       // a_b already folded into s_i
            e = (t > 0.f) ? t : ALPHA * t;      // leaky ReLU
        } else {
            e = NEG_INF;
        }
        // wave-wide max
        float m = e;
#pragma unroll
        for (int off = 16; off > 0; off >>= 1) m = fmaxf(m, __shfl_xor(m, off, 32));
        float p = __expf(e - m);
        float s = p;
#pragma unroll
        for (int off = 16; off > 0; off >>= 1) s += __shfl_xor(s, off, 32);
        const float attn = p / s;               // lane's coefficient for ctx slot `lane`

        // weighted sum of gathered rows: lane covers 4 contiguous features
        float a0 = 0.f, a1 = 0.f, a2 = 0.f, a3 = 0.f;
#pragma unroll 4
        for (int c = 0; c < NCTX; ++c) {
            const float w = __shfl(attn, c, 32);
            const int jc = nbr[row][c];
            if (jc >= 0) {
                const float* hp = h + (size_t)jc * IN_F + lane * 4;  // 512B coalesced, L2-hot
                a0 = fmaf(w, hp[0], a0);
                a1 = fmaf(w, hp[1], a1);
                a2 = fmaf(w, hp[2], a2);
                a3 = fmaf(w, hp[3], a3);
            }
        }
        mixed[row][lane * 4 + 0] = a0;
        mixed[row][lane * 4 + 1] = a1;
        mixed[row][lane * 4 + 2] = a2;
        mixed[row][lane * 4 + 3] = a3;
    }
    __syncthreads();

    // ---- Phase 2: fp32 WMMA GEMM  (exact fp32, K=128 in 32 steps of 4) ---
    const int colBase = wave * 16;       // wave's 16 output columns
    const int mrow = lane & 15;          // A: M index / B,D: N index
    const int khalf = (lane >> 4) * 2;   // lanes 0-15 -> K {k,k+1}; 16-31 -> {k+2,k+3}

    v8f acc = {};
#pragma unroll
    for (int k = 0; k < HID; k += 4) {
        v2f afrag, bfrag;
        // A (16x4, row-major in LDS): VGPR0=K(k+khalf), VGPR1=K(k+khalf+1)
        afrag.x = mixed[mrow][k + khalf];
        afrag.y = mixed[mrow][k + khalf + 1];
        // B (4x16 slice of W_j): row = K, col = colBase + mrow
        bfrag.x = Wj[(k + khalf) * HID + colBase + mrow];
        bfrag.y = Wj[(k + khalf + 1) * HID + colBase + mrow];
        acc = __builtin_amdgcn_wmma_f32_16x16x4_f32(
            /*neg_a=*/false, afrag, /*neg_b=*/false, bfrag,
            /*c_mod=*/(short)0, acc, /*reuse_a=*/false, /*reuse_b=*/false);
    }

    // D layout: VGPR r -> row r (lanes 0-15) / row r+8 (lanes 16-31)
    const int rbase = (lane >> 4) * 8;
#pragma unroll
    for (int r = 0; r < 8; ++r) {
        const int gr = tile * 16 + rbase + r;
        if (gr < N) out[(size_t)gr * HID + colBase + mrow] = acc[r];
    }
}

// ---------------------------------------------------------------------------
extern "C" void kernel_launch(void* const* d_in, const int* in_sizes, int n_in,
                              void* d_out, int out_size, void* d_ws, size_t ws_size,
                              hipStream_t stream) {
    const float*     h   = (const float*)d_in[0];
    const long long* idx = (const long long*)d_in[1];   // int64 indices
    const float*     Wi  = (const float*)d_in[2];
    const float*     Wj  = (const float*)d_in[3];
    const float*     aw  = (const float*)d_in[4];
    const float*     ab  = (const float*)d_in[5];
    float*           out = (float*)d_out;

    const int N = in_sizes[0] / IN_F;

    // scratch layout (floats): u[128] | v[128] | s_i[N] | s_all[N]
    float* u     = (float*)d_ws;
    float* v     = u + HID;
    float* s_i   = v + HID;
    float* s_all = s_i + N;

    gat_uv<<<1, 128, 0, stream>>>(Wi, Wj, aw, u, v);

    const int waves_blocks = (N * 32 + 255) / 256;
    gat_scores<<<waves_blocks, 256, 0, stream>>>(h, u, v, ab, s_i, s_all, N);

    const int tiles = (N + 15) / 16;
    gat_main<<<tiles, 256, 0, stream>>>(h, idx, Wj, s_i, s_all, out, N);
}